// DualPFE_81741817578051
// MI455X (gfx1250) — compile-verified
//
#include <hip/hip_runtime.h>
#include <hip/hip_bf16.h>

// ---------------------------------------------------------------------------
// DualPFE for MI455X (gfx1250, wave32).
// Problem sizes (from reference):
//   B=4, PB=12000, P=48000 pillars, N=32 pts, 8 raw feats, F=64
//   grid NY=496, NX=432; out = (4, 64, 496, 432) f32
// Strategy: one wave per pillar; both per-pillar GEMMs via
// V_WMMA_F32_16X16X4_F32 (f32, no precision loss). Output is zeroed and the
// blended feature scattered directly (empty BEV cells are exactly zero, and
// per-batch lin indices are unique -> no atomics, no dense BEV grids).
// ---------------------------------------------------------------------------

typedef __attribute__((ext_vector_type(2))) float v2f;
typedef __attribute__((ext_vector_type(8))) float v8f;

#define PTOT  48000
#define NPT   32
#define FD    64
#define NXg   432
#define NYg   496
#define EPSc  1e-3f

// -------------------------- kernel 1: pillar feature net -------------------
__global__ __launch_bounds__(256) void pfn_kernel(
    const float* __restrict__ pillars,   // (P, 32, 8)
    const float* __restrict__ w1,        // (10, 64)
    const float* __restrict__ bn1,       // (4, 64)  g,b,m,v
    const float* __restrict__ w2,        // (8, 64)
    const float* __restrict__ bn2,       // (4, 64)
    const int*   __restrict__ coords,    // (P, 4)
    const int*   __restrict__ nump,      // (P,)
    float* __restrict__ f1o,             // (P, 64)
    float* __restrict__ f2o)             // (P, 64)
{
    // per-wave staging: [0..9] pv_in, [10..11] zero pad (K=12), [12..19] raw
    __shared__ float lds[8][NPT][20];
    const int wave = threadIdx.x >> 5;
    const int lane = threadIdx.x & 31;
    const int lcol = lane & 15;
    const int hi   = lane >> 4;                 // 0: lanes 0-15, 1: lanes 16-31
    const int p    = blockIdx.x * 8 + wave;     // grid is exact: P/8 blocks

    // ---- each lane owns one point: load its 8 features ----
    const float* pp = pillars + ((size_t)p * NPT + lane) * 8;
    float x  = pp[0], y  = pp[1], z  = pp[2], i3 = pp[3];
    float i4 = pp[4], i5 = pp[5], i6 = pp[6], i7 = pp[7];

    // ---- cluster mean: sum xyz over all 32 points, / num_points ----
    float sx = x, sy = y, sz = z;
    #pragma unroll
    for (int off = 16; off >= 1; off >>= 1) {
        sx += __shfl_xor(sx, off, 32);
        sy += __shfl_xor(sy, off, 32);
        sz += __shfl_xor(sz, off, 32);
    }
    const float inv = 1.0f / (float)nump[p];
    const float mx = sx * inv, my = sy * inv, mz = sz * inv;

    const int c1 = coords[p * 4 + 1], c2 = coords[p * 4 + 2], c3 = coords[p * 4 + 3];
    const float cx = (float)c3 * 0.16f + 0.08f;
    const float cy = (float)c2 * 0.16f - 39.6f;
    const float cz = (float)c1 * 4.0f  - 1.0f;

    float* q = lds[wave][lane];
    q[0] = x;      q[1] = y;      q[2] = z;      q[3] = i3;
    q[4] = x - mx; q[5] = y - my; q[6] = z - mz;
    q[7] = x - cx; q[8] = y - cy; q[9] = z - cz;
    q[10] = 0.0f;  q[11] = 0.0f;
    q[12] = x;  q[13] = y;  q[14] = z;  q[15] = i3;
    q[16] = i4; q[17] = i5; q[18] = i6; q[19] = i7;
    __syncthreads();

    // ------------------- PFN1: [32 x 10(pad 12)] @ w1[10 x 64] -------------
    #pragma unroll
    for (int nt = 0; nt < 4; ++nt) {
        const int col = nt * 16 + lcol;
        v8f acc0 = {};   // points 0..15
        v8f acc1 = {};   // points 16..31
        #pragma unroll
        for (int kk = 0; kk < 3; ++kk) {
            const int c0 = kk * 4 + (hi ? 2 : 0);   // this half-wave's K pair
            v2f a0, a1, b;
            a0.x = lds[wave][lcol][c0];      a0.y = lds[wave][lcol][c0 + 1];
            a1.x = lds[wave][16 + lcol][c0]; a1.y = lds[wave][16 + lcol][c0 + 1];
            b.x = (c0     < 10) ? w1[c0 * FD + col]       : 0.0f;  // K pad rows
            b.y = (c0 + 1 < 10) ? w1[(c0 + 1) * FD + col] : 0.0f;
            acc0 = __builtin_amdgcn_wmma_f32_16x16x4_f32(false, a0, false, b,
                                                         (short)0, acc0, false, false);
            acc1 = __builtin_amdgcn_wmma_f32_16x16x4_f32(false, a1, false, b,
                                                         (short)0, acc1, false, false);
        }
        const float g  = bn1[0 * FD + col], bt = bn1[1 * FD + col];
        const float m  = bn1[2 * FD + col], vv = bn1[3 * FD + col];
        const float sc = g * rsqrtf(vv + EPSc);
        float h = -3.4e38f;
        #pragma unroll
        for (int i = 0; i < 8; ++i) {
            float h0 = (acc0[i] - m) * sc + bt;
            float h1 = (acc1[i] - m) * sc + bt;
            h = fmaxf(h, fmaxf(h0, h1));
        }
        h = fmaxf(h, 0.0f);                       // relu (commutes with max)
        h = fmaxf(h, __shfl_xor(h, 16, 32));      // merge rows M and M+8
        if (hi == 0) f1o[(size_t)p * FD + col] = h;
    }

    // ------------------- PFN2: [32 x 8] @ w2[8 x 64] -----------------------
    #pragma unroll
    for (int nt = 0; nt < 4; ++nt) {
        const int col = nt * 16 + lcol;
        v8f acc0 = {};
        v8f acc1 = {};
        #pragma unroll
        for (int kk = 0; kk < 2; ++kk) {
            const int r0 = kk * 4 + (hi ? 2 : 0);
            const int c0 = 12 + r0;               // raw feats at LDS offset 12
            v2f a0, a1, b;
            a0.x = lds[wave][lcol][c0];      a0.y = lds[wave][lcol][c0 + 1];
            a1.x = lds[wave][16 + lcol][c0]; a1.y = lds[wave][16 + lcol][c0 + 1];
            b.x = w2[r0 * FD + col];
            b.y = w2[(r0 + 1) * FD + col];
            acc0 = __builtin_amdgcn_wmma_f32_16x16x4_f32(false, a0, false, b,
                                                         (short)0, acc0, false, false);
            acc1 = __builtin_amdgcn_wmma_f32_16x16x4_f32(false, a1, false, b,
                                                         (short)0, acc1, false, false);
        }
        const float g  = bn2[0 * FD + col], bt = bn2[1 * FD + col];
        const float m  = bn2[2 * FD + col], vv = bn2[3 * FD + col];
        const float sc = g * rsqrtf(vv + EPSc);
        float h = -3.4e38f;
        #pragma unroll
        for (int i = 0; i < 8; ++i) {
            float h0 = (acc0[i] - m) * sc + bt;
            float h1 = (acc1[i] - m) * sc + bt;
            h = fmaxf(h, fmaxf(h0, h1));
        }
        h = fmaxf(h, 0.0f);
        h = fmaxf(h, __shfl_xor(h, 16, 32));
        if (hi == 0) f2o[(size_t)p * FD + col] = h;
    }
}

// -------------------------- kernel 2: zero output --------------------------
__global__ __launch_bounds__(256) void zero_kernel(float4* __restrict__ out, size_t n4)
{
    size_t i = (size_t)blockIdx.x * blockDim.x + threadIdx.x;
    if (i < n4) out[i] = make_float4(0.f, 0.f, 0.f, 0.f);
}

// ------------------- kernel 3: attention blend + scatter -------------------
__global__ __launch_bounds__(256) void fuse_kernel(
    const float* __restrict__ f1,     // (P, 64)
    const float* __restrict__ f2,     // (P, 64)
    const float* __restrict__ wf1,    // (64,)
    const float* __restrict__ wf2,    // (64,)
    const float* __restrict__ fbn1,   // (4,)  g,b,m,v
    const float* __restrict__ fbn2,   // (4,)
    const int*   __restrict__ coords, // (P, 4)
    float* __restrict__ out)          // (B, 64, NY, NX)
{
    const int wave = threadIdx.x >> 5;
    const int lane = threadIdx.x & 31;
    const int p    = blockIdx.x * 8 + wave;

    const float* F1 = f1 + (size_t)p * FD;
    const float* F2 = f2 + (size_t)p * FD;
    const int fA = 2 * lane, fB = 2 * lane + 1;

    float s1 = F1[fA] * wf1[fA] + F1[fB] * wf1[fB];
    float s2 = F2[fA] * wf2[fA] + F2[fB] * wf2[fB];
    #pragma unroll
    for (int off = 16; off >= 1; off >>= 1) {
        s1 += __shfl_xor(s1, off, 32);
        s2 += __shfl_xor(s2, off, 32);
    }
    s1 = (s1 - fbn1[2]) * (fbn1[0] * rsqrtf(fbn1[3] + EPSc)) + fbn1[1];
    s2 = (s2 - fbn2[2]) * (fbn2[0] * rsqrtf(fbn2[3] + EPSc)) + fbn2[1];
    const float mm = fmaxf(s1, s2);
    const float e1 = __expf(s1 - mm), e2 = __expf(s2 - mm);
    const float rs = 1.0f / (e1 + e2);
    const float a1 = e1 * rs, a2 = e2 * rs;

    const int b  = coords[p * 4 + 0];
    const int yy = coords[p * 4 + 2];
    const int xx = coords[p * 4 + 3];
    const size_t plane = (size_t)NYg * NXg;
    const size_t base  = ((size_t)b * FD) * plane + (size_t)yy * NXg + xx;
    out[base + (size_t)fA * plane] = F1[fA] * a1 + F2[fA] * a2;
    out[base + (size_t)fB * plane] = F1[fB] * a1 + F2[fB] * a2;
}

// ---------------------------------------------------------------------------
extern "C" void kernel_launch(void* const* d_in, const int* in_sizes, int n_in,
                              void* d_out, int out_size, void* d_ws, size_t ws_size,
                              hipStream_t stream) {
    const float* pillars = (const float*)d_in[0];
    const float* w1      = (const float*)d_in[1];
    const float* bn1     = (const float*)d_in[2];
    const float* w2      = (const float*)d_in[3];
    const float* bn2     = (const float*)d_in[4];
    const float* wf1     = (const float*)d_in[5];
    const float* wf2     = (const float*)d_in[6];
    const float* fbn1    = (const float*)d_in[7];
    const float* fbn2    = (const float*)d_in[8];
    const int*   coords  = (const int*)d_in[9];
    const int*   nump    = (const int*)d_in[10];
    float* out = (float*)d_out;

    float* f1 = (float*)d_ws;                    // P*64 floats (12.3 MB)
    float* f2 = f1 + (size_t)PTOT * FD;          // P*64 floats (12.3 MB)

    pfn_kernel<<<PTOT / 8, 256, 0, stream>>>(pillars, w1, bn1, w2, bn2,
                                             coords, nump, f1, f2);

    const size_t n4 = (size_t)out_size / 4;      // out_size divisible by 4
    zero_kernel<<<(unsigned)((n4 + 255) / 256), 256, 0, stream>>>((float4*)out, n4);

    fuse_kernel<<<PTOT / 8, 256, 0, stream>>>(f1, f2, wf1, wf2, fbn1, fbn2,
                                              coords, out);
}